// SimpleMinGRU_5626407158120
// MI455X (gfx1250) — compile-verified
//
#include <hip/hip_runtime.h>
#include <hip/hip_bf16.h>

// SimpleMinGRU fused pipeline for MI455X (gfx1250, wave32).
//   K1: fused input-proj + dual GEMM (Wz, Wh) via V_WMMA_F32_16X16X4_F32,
//       128-row M-tiles so Wz/Wh L2 traffic is 256MB (not 2.1GB).
//   K2a/2b/2c: chunked parallel linear scan over L (h_t = a_t*h_{t-1} + b_t)
//   K3: readout GEMV with wave32 shuffle reduction (uses h_{t-1})
// Bandwidth-bound (~800MB HBM + 256MB L2); fp32 WMMA keeps reference precision.

#define BB     64
#define LL     4096
#define HH     128
#define DIN    3
#define ROWS   128                 // M rows per block in K1
#define MT     (ROWS / 16)         // 8 M-subtiles per wave
#define NCHUNK 32
#define TCHUNK (LL / NCHUNK)       // 128

typedef __attribute__((ext_vector_type(2))) float v2f;
typedef __attribute__((ext_vector_type(8))) float v8f;

__device__ __forceinline__ float fast_sigmoid(float s) {
    return 1.0f / (1.0f + __expf(-s));
}
__device__ __forceinline__ float fast_tanh(float s) {
    // tanh(x) = 1 - 2/(exp(2x)+1); exact limits at +-inf, monotone.
    return 1.0f - 2.0f / (__expf(2.0f * s) + 1.0f);
}

// ---------------------------------------------------------------------------
// Kernel 1: for ROWS rows of (B*L): inp = x@Wp+bp (K=3, computed into LDS),
// then Sz = inp@Wz+bz, Sh = inp@Wh+bh via fp32 WMMA; emit a=1-sigmoid(Sz),
// b=sigmoid(Sz)*tanh(Sh).
// Block = 256 threads = 8 waves; wave w owns output columns [16w,16w+16) and
// iterates MT=8 M-subtiles per k-step, reusing each B fragment 8x.
// ---------------------------------------------------------------------------
__global__ __launch_bounds__(256) void mingru_gates_wmma(
    const float* __restrict__ x,    // (B*L, 3)
    const float* __restrict__ Wp,   // (3, H)
    const float* __restrict__ bp,   // (H)
    const float* __restrict__ Wz,   // (H, H) row-major (k, n)
    const float* __restrict__ bz,   // (H)
    const float* __restrict__ Wh,   // (H, H)
    const float* __restrict__ bh,   // (H)
    float* __restrict__ a_out,      // (B*L, H)
    float* __restrict__ b_out)      // (B*L, H)
{
    __shared__ float lds_inp[ROWS * HH];   // 64 KB fp32 A-tile
    __shared__ float lds_x[ROWS * 4];      // staged x rows (padded to 4)

    const int  tid  = threadIdx.x;
    const long row0 = (long)blockIdx.x * ROWS;

    for (int i = tid; i < ROWS * DIN; i += 256) {
        const int m = i / DIN, d = i % DIN;
        lds_x[m * 4 + d] = x[(row0 + m) * DIN + d];
    }
    __syncthreads();

    // inp tile: ROWSx128, fused K=3 projection
    for (int i = tid; i < ROWS * HH; i += 256) {
        const int m = i >> 7;
        const int c = i & (HH - 1);
        float v = bp[c];
        v += lds_x[m * 4 + 0] * Wp[0 * HH + c];
        v += lds_x[m * 4 + 1] * Wp[1 * HH + c];
        v += lds_x[m * 4 + 2] * Wp[2 * HH + c];
        lds_inp[i] = v;
    }
    __syncthreads();

    const int wave  = tid >> 5;          // 0..7 -> column tile
    const int lane  = tid & 31;
    const int lhalf = lane >> 4;         // 0: K pair (k0,k0+1) / 1: (k0+2,k0+3)
    const int l16   = lane & 15;
    const int ncol  = wave * 16 + l16;   // this lane's output column

    v8f accz[MT];
    v8f acch[MT];
    #pragma unroll
    for (int mt = 0; mt < MT; ++mt) { accz[mt] = (v8f){}; acch[mt] = (v8f){}; }

    #pragma unroll 4
    for (int k0 = 0; k0 < HH; k0 += 4) {
        const int ka = k0 + lhalf * 2;
        // B 4x16 fp32 frags: lane -> N=ncol, K={ka, ka+1}; reused by MT wmmas
        v2f bzf, bhf;
        bzf.x = Wz[(ka    ) * HH + ncol];
        bzf.y = Wz[(ka + 1) * HH + ncol];
        bhf.x = Wh[(ka    ) * HH + ncol];
        bhf.y = Wh[(ka + 1) * HH + ncol];
        #pragma unroll
        for (int mt = 0; mt < MT; ++mt) {
            // A 16x4 fp32 frag: lane -> M=l16 (of subtile mt), K={ka, ka+1}
            v2f afrag;
            afrag.x = lds_inp[(mt * 16 + l16) * HH + ka];
            afrag.y = lds_inp[(mt * 16 + l16) * HH + ka + 1];
            accz[mt] = __builtin_amdgcn_wmma_f32_16x16x4_f32(
                false, afrag, false, bzf, (short)0, accz[mt], false, false);
            acch[mt] = __builtin_amdgcn_wmma_f32_16x16x4_f32(
                false, afrag, false, bhf, (short)0, acch[mt], false, false);
        }
    }

    const float biasz = bz[ncol];
    const float biash = bh[ncol];
    #pragma unroll
    for (int mt = 0; mt < MT; ++mt) {
        #pragma unroll
        for (int v = 0; v < 8; ++v) {
            const long row = row0 + mt * 16 + lhalf * 8 + v;  // lanes>=16: M=8..15
            const float z  = fast_sigmoid(accz[mt][v] + biasz);
            const float ht = fast_tanh(acch[mt][v] + biash);
            a_out[row * HH + ncol] = 1.0f - z;
            b_out[row * HH + ncol] = z * ht;
        }
    }
}

// ---------------------------------------------------------------------------
// Kernel 2a: per-chunk composition. thread = (b, chunk, h).
// (A,B) o (a_t,b_t) = (A*a_t, B*a_t + b_t), identity (1,0).
// ---------------------------------------------------------------------------
__global__ __launch_bounds__(128) void scan_chunk_reduce(
    const float* __restrict__ a_arr, const float* __restrict__ b_arr,
    float* __restrict__ aggA, float* __restrict__ aggB)   // (B*NCHUNK, H)
{
    const int  bidx = blockIdx.x;               // b*NCHUNK + c
    const int  b    = bidx / NCHUNK;
    const int  c    = bidx % NCHUNK;
    const int  h    = threadIdx.x;
    const long base = ((long)b * LL + (long)c * TCHUNK) * HH + h;

    float A = 1.0f, Bv = 0.0f;
    for (int t0 = 0; t0 < TCHUNK; t0 += 16) {
        __builtin_prefetch(&a_arr[base + (long)(t0 + 16) * HH], 0, 0);
        __builtin_prefetch(&b_arr[base + (long)(t0 + 16) * HH], 0, 0);
        #pragma unroll
        for (int t = 0; t < 16; ++t) {
            const long off = base + (long)(t0 + t) * HH;
            const float a  = a_arr[off];
            const float bb = b_arr[off];
            Bv = Bv * a + bb;
            A  = A * a;
        }
    }
    aggA[(long)bidx * HH + h] = A;
    aggB[(long)bidx * HH + h] = Bv;
}

// ---------------------------------------------------------------------------
// Kernel 2b: scan the NCHUNK aggregates per (b,h) -> h at each chunk start.
// ---------------------------------------------------------------------------
__global__ __launch_bounds__(128) void scan_chunk_prefix(
    const float* __restrict__ aggA, const float* __restrict__ aggB,
    float* __restrict__ hinit)                  // (B*NCHUNK, H)
{
    const int b = blockIdx.x;
    const int h = threadIdx.x;
    float s = 0.0f;
    #pragma unroll
    for (int c = 0; c < NCHUNK; ++c) {
        const long idx = ((long)b * NCHUNK + c) * HH + h;
        hinit[idx] = s;
        s = aggA[idx] * s + aggB[idx];
    }
}

// ---------------------------------------------------------------------------
// Kernel 2c: replay each chunk from its prefix, writing h IN PLACE over a_arr
// (valid: a[t] is read before h[t] overwrites it; deterministic because K1
// regenerates a_arr on every launch).
// ---------------------------------------------------------------------------
__global__ __launch_bounds__(128) void scan_chunk_apply(
    float* __restrict__ a_arr,                  // in: a, out: h
    const float* __restrict__ b_arr,
    const float* __restrict__ hinit)
{
    const int  bidx = blockIdx.x;
    const int  b    = bidx / NCHUNK;
    const int  c    = bidx % NCHUNK;
    const int  h    = threadIdx.x;
    const long base = ((long)b * LL + (long)c * TCHUNK) * HH + h;

    float hp = hinit[(long)bidx * HH + h];
    for (int t0 = 0; t0 < TCHUNK; t0 += 16) {
        __builtin_prefetch(&a_arr[base + (long)(t0 + 16) * HH], 0, 0);
        __builtin_prefetch(&b_arr[base + (long)(t0 + 16) * HH], 0, 0);
        #pragma unroll
        for (int t = 0; t < 16; ++t) {
            const long off = base + (long)(t0 + t) * HH;
            const float a  = a_arr[off];
            const float bb = b_arr[off];
            hp = a * hp + bb;
            a_arr[off] = hp;
        }
    }
}

// ---------------------------------------------------------------------------
// Kernel 3: preds[o] = bg + Wg . h[o-1]   (o = b*L + t; t==0 -> bg only).
// One wave32 per output; float4 per lane + shuffle-xor reduction.
// ---------------------------------------------------------------------------
__global__ __launch_bounds__(256) void readout_gemv(
    const float* __restrict__ h_arr, const float* __restrict__ Wg,
    const float* __restrict__ bg, float* __restrict__ preds)
{
    const int  wave = threadIdx.x >> 5;
    const int  lane = threadIdx.x & 31;
    const long o    = (long)blockIdx.x * 8 + wave;
    const long t    = o & (LL - 1);

    float sum = 0.0f;
    if (t != 0) {                                // wave-uniform branch
        const float4 hv = *(const float4*)(h_arr + (o - 1) * HH + lane * 4);
        const float4 wv = *(const float4*)(Wg + lane * 4);
        sum = hv.x * wv.x + hv.y * wv.y + hv.z * wv.z + hv.w * wv.w;
    }
    #pragma unroll
    for (int off = 16; off > 0; off >>= 1)
        sum += __shfl_xor(sum, off, 32);
    if (lane == 0)
        preds[o] = sum + bg[0];
}

// ---------------------------------------------------------------------------
extern "C" void kernel_launch(void* const* d_in, const int* in_sizes, int n_in,
                              void* d_out, int out_size, void* d_ws, size_t ws_size,
                              hipStream_t stream) {
    const float* x  = (const float*)d_in[0];
    const float* Wp = (const float*)d_in[1];
    const float* bp = (const float*)d_in[2];
    const float* Wz = (const float*)d_in[3];
    const float* bz = (const float*)d_in[4];
    const float* Wh = (const float*)d_in[5];
    const float* bh = (const float*)d_in[6];
    const float* Wg = (const float*)d_in[7];
    const float* bg = (const float*)d_in[8];
    float* preds = (float*)d_out;

    const size_t BLH = (size_t)BB * LL * HH;        // 33,554,432 floats
    const size_t AGG = (size_t)BB * NCHUNK * HH;    // 262,144 floats
    float* ws    = (float*)d_ws;
    float* a_arr = ws;                               // later overwritten with h
    float* b_arr = ws + BLH;
    float* aggA  = ws + 2 * BLH;
    float* aggB  = aggA + AGG;
    float* hinit = aggB + AGG;

    // K1: gates via fp32 WMMA. 2048 row-tiles of 128.
    mingru_gates_wmma<<<(BB * LL) / ROWS, 256, 0, stream>>>(
        x, Wp, bp, Wz, bz, Wh, bh, a_arr, b_arr);

    // K2: chunked scan (2048-way -> 64-way -> 2048-way parallel passes).
    scan_chunk_reduce<<<BB * NCHUNK, HH, 0, stream>>>(a_arr, b_arr, aggA, aggB);
    scan_chunk_prefix<<<BB, HH, 0, stream>>>(aggA, aggB, hinit);
    scan_chunk_apply <<<BB * NCHUNK, HH, 0, stream>>>(a_arr, b_arr, hinit);

    // K3: readout (a_arr now holds h).
    readout_gemv<<<(BB * LL) / 8, 256, 0, stream>>>(a_arr, Wg, bg, preds);
}